// LinearTransformer_75892072120460
// MI455X (gfx1250) — compile-verified
//
#include <hip/hip_runtime.h>
#include <hip/hip_bf16.h>

// ---------------------------------------------------------------------------
// LinearTransformer forward for MI455X (gfx1250, wave32, WMMA).
// - All GEMMs: V_WMMA_F32_16X16X4_F32, 16x64 output per wave (4 accumulators).
// - B tiles staged via GLOBAL_LOAD_ASYNC_TO_LDS_B128, double-buffered,
//   overlapped with WMMA compute (ASYNCcnt + one barrier per K-chunk).
// ---------------------------------------------------------------------------

typedef __attribute__((ext_vector_type(2))) float v2f;
typedef __attribute__((ext_vector_type(8))) float v8f;

#define NB      16
#define LSEQ    4096
#define DMODEL  512
#define INDIM   32
#define MTOT    (NB * LSEQ)          // 65536 rows total
#define EPS_ATTN 1e-6f
#define EPS_LN   1e-5f
#define CHPAD   132                  // padded K-stride for W^T LDS tiles

// D = A(16x4) * B(4x16) + C, fp32, wave32.
__device__ __forceinline__ v8f wmma4(v2f a, v2f b, v8f c) {
    return __builtin_amdgcn_wmma_f32_16x16x4_f32(
        /*neg_a=*/false, a, /*neg_b=*/false, b,
        /*c_mod=*/(short)0, c, /*reuse_a=*/false, /*reuse_b=*/false);
}

// Async global->LDS copy, 16 bytes per active lane (ASYNCcnt-tracked).
__device__ __forceinline__ void async_b128(void* lds, const void* gp) {
    const unsigned l = (unsigned)(uintptr_t)lds;   // addr[31:0] == LDS offset
    asm volatile("global_load_async_to_lds_b128 %0, %1, off"
                 :: "v"(l), "v"(gp) : "memory");
}
__device__ __forceinline__ void wait_async0() {
    asm volatile("s_wait_asynccnt 0" ::: "memory");
}

__device__ __forceinline__ float wred(float s) {
#pragma unroll
    for (int off = 16; off; off >>= 1) s += __shfl_xor(s, off, 32);
    return s;
}

// ---------------------------------------------------------------------------
// out[M,Nout] = act( A[M,Kd] @ W[Nout,Kd]^T + bias ) (+ residual)
// ACT: 0 none, 1 elu+1 (x>0 ? x+1 : exp(x)), 2 relu
// Block: 256 thr = 8 waves; block tile 128 rows x 64 cols (wave: 16x64).
// LDS B layout: Bs[n*CHPAD + k] (k contiguous, matches W rows for b128 copies)
// ---------------------------------------------------------------------------
template <int ACT, bool RES>
__global__ void __launch_bounds__(256)
gemm_wT(const float* __restrict__ A, const float* __restrict__ W,
        const float* __restrict__ bias, const float* __restrict__ res,
        float* __restrict__ out, int M, int Kd, int Nout) {
    __shared__ float Bs[2][64 * CHPAD];
    const int lane = threadIdx.x & 31;
    const int wv   = threadIdx.x >> 5;
    const int lo   = lane & 15;
    const int hi   = lane >> 4;                 // selects K pair / M+8
    const int col0 = blockIdx.x * 64;
    const int row0 = blockIdx.y * 128 + wv * 16;

    v8f acc[4];
#pragma unroll
    for (int j = 0; j < 4; ++j) {
        const float bv = bias[col0 + 16 * j + lo];
#pragma unroll
        for (int r = 0; r < 8; ++r) acc[j][r] = bv;
    }

    const int nch = (Kd + 127) >> 7;
    {   // prologue: async-stage chunk 0
        const int ch4 = (Kd < 128 ? Kd : 128) >> 2;
        for (int i = threadIdx.x; i < 64 * ch4; i += 256) {
            const int n = i / ch4, k4 = i % ch4;
            async_b128(&Bs[0][n * CHPAD + k4 * 4],
                       W + (size_t)(col0 + n) * Kd + k4 * 4);
        }
    }
    wait_async0();
    __syncthreads();

    for (int c = 0; c < nch; ++c) {
        const int k0 = c << 7;
        const int ch = (Kd - k0) < 128 ? (Kd - k0) : 128;
        const float* cur = Bs[c & 1];
        if (c + 1 < nch) {                       // overlap next-chunk copy
            const int k0n  = k0 + 128;
            const int chn4 = ((Kd - k0n) < 128 ? (Kd - k0n) : 128) >> 2;
            float* nxt = Bs[(c + 1) & 1];
            for (int i = threadIdx.x; i < 64 * chn4; i += 256) {
                const int n = i / chn4, k4 = i % chn4;
                async_b128(&nxt[n * CHPAD + k4 * 4],
                           W + (size_t)(col0 + n) * Kd + k0n + k4 * 4);
            }
        }
        const float* Arow = A + (size_t)(row0 + lo) * Kd + k0;
        __builtin_prefetch(Arow + 128, 0, 1);
        for (int k = 0; k < ch; k += 4) {
            const int kk = k + 2 * hi;
            const v2f a = *(const v2f*)(Arow + kk);     // A[m][kk..kk+1]
#pragma unroll
            for (int j = 0; j < 4; ++j) {
                const v2f b = *(const v2f*)(cur + (lo + 16 * j) * CHPAD + kk);
                acc[j] = wmma4(a, b, acc[j]);
            }
        }
        wait_async0();
        __syncthreads();
    }

#pragma unroll
    for (int j = 0; j < 4; ++j) {
#pragma unroll
        for (int r = 0; r < 8; ++r) {
            const int row = row0 + r + 8 * hi;
            const int col = col0 + 16 * j + lo;
            float v = acc[j][r];
            if (ACT == 1) v = v > 0.f ? v + 1.f : __expf(v);  // elu(x)+1
            if (ACT == 2) v = v > 0.f ? v : 0.f;
            if (RES) v += res[(size_t)row * Nout + col];
            out[(size_t)row * Nout + col] = v;
        }
    }
}

// ---------------------------------------------------------------------------
// KV[n][e][m] = sum_l K[n][l][e] * V[n][l][m]  (per-batch K^T @ V, Kdim = L)
// grid = (DMODEL/64, DMODEL/128, NB). LDS B layout: Bs[l*64 + m] (m contig).
// ---------------------------------------------------------------------------
__global__ void __launch_bounds__(256)
kv_gemm(const float* __restrict__ Kf, const float* __restrict__ V,
        float* __restrict__ KV) {
    __shared__ float Bs[2][128 * 64];
    const int lane = threadIdx.x & 31;
    const int wv   = threadIdx.x >> 5;
    const int lo   = lane & 15;
    const int hi   = lane >> 4;
    const int n    = blockIdx.z;
    const int col0 = blockIdx.x * 64;                 // m
    const int row0 = blockIdx.y * 128 + wv * 16;      // e

    const float* Kn = Kf + (size_t)n * LSEQ * DMODEL;
    const float* Vn = V  + (size_t)n * LSEQ * DMODEL;

    v8f acc[4];
#pragma unroll
    for (int j = 0; j < 4; ++j)
#pragma unroll
        for (int r = 0; r < 8; ++r) acc[j][r] = 0.f;

    // prologue: stage V chunk 0
    for (int i = threadIdx.x; i < 128 * 16; i += 256) {
        const int l = i >> 4, m4 = i & 15;
        async_b128(&Bs[0][l * 64 + m4 * 4],
                   Vn + (size_t)l * DMODEL + col0 + m4 * 4);
    }
    wait_async0();
    __syncthreads();

    const int nch = LSEQ / 128;
    for (int c = 0; c < nch; ++c) {
        const int l0 = c << 7;
        const float* cur = Bs[c & 1];
        if (c + 1 < nch) {
            float* nxt = Bs[(c + 1) & 1];
            const float* Vsrc = Vn + (size_t)(l0 + 128) * DMODEL;
            for (int i = threadIdx.x; i < 128 * 16; i += 256) {
                const int l = i >> 4, m4 = i & 15;
                async_b128(&nxt[l * 64 + m4 * 4],
                           Vsrc + (size_t)l * DMODEL + col0 + m4 * 4);
            }
        }
#pragma unroll 4
        for (int l = 0; l < 128; l += 4) {
            const int ll = l + 2 * hi;
            v2f a;                                        // A[e][l] = K[l][e]
            a.x = Kn[(size_t)(l0 + ll) * DMODEL + row0 + lo];
            a.y = Kn[(size_t)(l0 + ll + 1) * DMODEL + row0 + lo];
#pragma unroll
            for (int j = 0; j < 4; ++j) {
                v2f b;
                b.x = cur[ll * 64 + lo + 16 * j];
                b.y = cur[(ll + 1) * 64 + lo + 16 * j];
                acc[j] = wmma4(a, b, acc[j]);
            }
        }
        wait_async0();
        __syncthreads();
    }

    float* KVn = KV + (size_t)n * DMODEL * DMODEL;
#pragma unroll
    for (int j = 0; j < 4; ++j)
#pragma unroll
        for (int r = 0; r < 8; ++r)
            KVn[(size_t)(row0 + r + 8 * hi) * DMODEL + col0 + 16 * j + lo] =
                acc[j][r];
}

// Ksum[n][e] = sum_l K[n][l][e]
__global__ void __launch_bounds__(256)
ksum_kernel(const float* __restrict__ Kf, float* __restrict__ Ksum) {
    const int n = blockIdx.y;
    const int e = blockIdx.x * 256 + threadIdx.x;
    const float* Kn = Kf + (size_t)n * LSEQ * DMODEL;
    float s = 0.f;
    for (int l = 0; l < LSEQ; ++l) s += Kn[(size_t)l * DMODEL + e];
    Ksum[n * DMODEL + e] = s;
}

// Z[n*L + l] = 1 / (dot(Q[n,l,:], Ksum[n,:]) + eps)   (one wave per row)
__global__ void __launch_bounds__(256)
z_kernel(const float* __restrict__ Q, const float* __restrict__ Ksum,
         float* __restrict__ Z) {
    const int lane = threadIdx.x & 31;
    const int row  = blockIdx.x * 8 + (threadIdx.x >> 5);
    const int n    = row >> 12;                         // row / LSEQ
    const float* q  = Q + (size_t)row * DMODEL;
    const float* ks = Ksum + n * DMODEL;
    float s = 0.f;
    for (int d = lane; d < DMODEL; d += 32) s += q[d] * ks[d];
    s = wred(s);
    if (lane == 0) Z[row] = 1.f / (s + EPS_ATTN);
}

// attnS[n,l,m] = Z[n,l] * sum_d Q[n,l,d] * KV[n][d][m]
// grid = (DMODEL/64, LSEQ/128, NB)
__global__ void __launch_bounds__(256)
attn_gemm(const float* __restrict__ Q, const float* __restrict__ KV,
          const float* __restrict__ Z, float* __restrict__ out) {
    __shared__ float Bs[2][128 * 64];
    const int lane = threadIdx.x & 31;
    const int wv   = threadIdx.x >> 5;
    const int lo   = lane & 15;
    const int hi   = lane >> 4;
    const int n    = blockIdx.z;
    const int col0 = blockIdx.x * 64;
    const int row0 = blockIdx.y * 128 + wv * 16;        // l within batch

    const float* Qn = Q  + (size_t)n * LSEQ * DMODEL;
    const float* Bn = KV + (size_t)n * DMODEL * DMODEL;

    v8f acc[4];
#pragma unroll
    for (int j = 0; j < 4; ++j)
#pragma unroll
        for (int r = 0; r < 8; ++r) acc[j][r] = 0.f;

    for (int i = threadIdx.x; i < 128 * 16; i += 256) {
        const int d = i >> 4, m4 = i & 15;
        async_b128(&Bs[0][d * 64 + m4 * 4],
                   Bn + (size_t)d * DMODEL + col0 + m4 * 4);
    }
    wait_async0();
    __syncthreads();

    const int nch = DMODEL / 128;
    for (int c = 0; c < nch; ++c) {
        const int d0 = c << 7;
        const float* cur = Bs[c & 1];
        if (c + 1 < nch) {
            float* nxt = Bs[(c + 1) & 1];
            const float* Bsrc = Bn + (size_t)(d0 + 128) * DMODEL;
            for (int i = threadIdx.x; i < 128 * 16; i += 256) {
                const int d = i >> 4, m4 = i & 15;
                async_b128(&nxt[d * 64 + m4 * 4],
                           Bsrc + (size_t)d * DMODEL + col0 + m4 * 4);
            }
        }
        const float* Qrow = Qn + (size_t)(row0 + lo) * DMODEL + d0;
        __builtin_prefetch(Qrow + 128, 0, 1);
#pragma unroll 4
        for (int d = 0; d < 128; d += 4) {
            const int dd = d + 2 * hi;
            const v2f a = *(const v2f*)(Qrow + dd);
#pragma unroll
            for (int j = 0; j < 4; ++j) {
                v2f b;
                b.x = cur[dd * 64 + lo + 16 * j];
                b.y = cur[(dd + 1) * 64 + lo + 16 * j];
                acc[j] = wmma4(a, b, acc[j]);
            }
        }
        wait_async0();
        __syncthreads();
    }

#pragma unroll
    for (int j = 0; j < 4; ++j)
#pragma unroll
        for (int r = 0; r < 8; ++r) {
            const int row = row0 + r + 8 * hi;
            const float z = Z[n * LSEQ + row];
            out[((size_t)n * LSEQ + row) * DMODEL + col0 + 16 * j + lo] =
                acc[j][r] * z;
        }
}

// y = LN(x) * g + b, one wave per 512-wide row
__global__ void __launch_bounds__(256)
ln_kernel(const float* __restrict__ x, const float* __restrict__ g,
          const float* __restrict__ b, float* __restrict__ y) {
    const int lane = threadIdx.x & 31;
    const size_t row = (size_t)blockIdx.x * 8 + (threadIdx.x >> 5);
    const float* xr = x + row * DMODEL;
    float lv[16];
    float s = 0.f;
#pragma unroll
    for (int i = 0; i < 16; ++i) { lv[i] = xr[lane + 32 * i]; s += lv[i]; }
    const float mu = wred(s) * (1.f / DMODEL);
    float v = 0.f;
#pragma unroll
    for (int i = 0; i < 16; ++i) { const float d = lv[i] - mu; v += d * d; }
    const float inv = rsqrtf(wred(v) * (1.f / DMODEL) + EPS_LN);
    float* yr = y + row * DMODEL;
#pragma unroll
    for (int i = 0; i < 16; ++i) {
        const int d = lane + 32 * i;
        yr[d] = (lv[i] - mu) * inv * g[d] + b[d];
    }
}

// Final: per batch n, row l=0: LN(g2,b2) -> LN(gf,bf) -> dot(w_fc)+b_fc
__global__ void __launch_bounds__(32)
final_kernel(const float* __restrict__ u2,
             const float* __restrict__ g2, const float* __restrict__ b2,
             const float* __restrict__ gf, const float* __restrict__ bf,
             const float* __restrict__ wfc, const float* __restrict__ bfc,
             float* __restrict__ out) {
    const int lane = threadIdx.x;
    const float* xr = u2 + (size_t)blockIdx.x * LSEQ * DMODEL;  // l = 0 row
    float lv[16];
    float s = 0.f;
#pragma unroll
    for (int i = 0; i < 16; ++i) { lv[i] = xr[lane + 32 * i]; s += lv[i]; }
    float mu = wred(s) * (1.f / DMODEL);
    float v = 0.f;
#pragma unroll
    for (int i = 0; i < 16; ++i) { const float d = lv[i] - mu; v += d * d; }
    float inv = rsqrtf(wred(v) * (1.f / DMODEL) + EPS_LN);
#pragma unroll
    for (int i = 0; i < 16; ++i) {
        const int d = lane + 32 * i;
        lv[i] = (lv[i] - mu) * inv * g2[d] + b2[d];
    }
    s = 0.f;
#pragma unroll
    for (int i = 0; i < 16; ++i) s += lv[i];
    mu = wred(s) * (1.f / DMODEL);
    v = 0.f;
#pragma unroll
    for (int i = 0; i < 16; ++i) { const float d = lv[i] - mu; v += d * d; }
    inv = rsqrtf(wred(v) * (1.f / DMODEL) + EPS_LN);
    float acc = 0.f;
#pragma unroll
    for (int i = 0; i < 16; ++i) {
        const int d = lane + 32 * i;
        const float hf = (lv[i] - mu) * inv * gf[d] + bf[d];
        acc += hf * wfc[d];
    }
    acc = wred(acc);
    if (lane == 0) out[blockIdx.x] = acc + bfc[0];
}

// ---------------------------------------------------------------------------
extern "C" void kernel_launch(void* const* d_in, const int* in_sizes, int n_in,
                              void* d_out, int out_size, void* d_ws, size_t ws_size,
                              hipStream_t stream) {
    (void)in_sizes; (void)n_in; (void)out_size; (void)ws_size;

    const float* x     = (const float*)d_in[0];
    const float* w_in  = (const float*)d_in[1];
    const float* b_in  = (const float*)d_in[2];
    const float* wq    = (const float*)d_in[3];
    const float* bq    = (const float*)d_in[4];
    const float* wk    = (const float*)d_in[5];
    const float* bk    = (const float*)d_in[6];
    const float* wvv   = (const float*)d_in[7];
    const float* bvv   = (const float*)d_in[8];
    const float* wo    = (const float*)d_in[9];
    const float* bo    = (const float*)d_in[10];
    const float* g1    = (const float*)d_in[11];
    const float* b1    = (const float*)d_in[12];
    const float* w_ff1 = (const float*)d_in[13];
    const float* b_ff1 = (const float*)d_in[14];
    const float* w_ff2 = (const float*)d_in[15];
    const float* b_ff2 = (const float*)d_in[16];
    const float* g2    = (const float*)d_in[17];
    const float* b2    = (const float*)d_in[18];
    const float* gf    = (const float*)d_in[19];
    const float* bf    = (const float*)d_in[20];
    const float* w_fc  = (const float*)d_in[21];
    const float* b_fc  = (const float*)d_in[22];

    float* ws = (float*)d_ws;
    const size_t S = (size_t)MTOT * DMODEL;       // 33,554,432 floats
    float* h    = ws;                             // [M,D]
    float* Qb   = ws + S;                         // [M,D]
    float* Kb   = ws + 2 * S;                     // [M,D]
    float* Vb   = ws + 3 * S;                     // [M,D]
    float* KVb  = ws + 4 * S;                     // [N,E,E]
    float* Ksum = KVb + (size_t)NB * DMODEL * DMODEL;  // [N,E]
    float* Zb   = Ksum + NB * DMODEL;             // [N*L]
    // aliases (lifetimes are disjoint along the stream):
    float* attnS = Vb;    // V dead after kv_gemm
    float* u1    = Kb;    // K dead after kv_gemm/ksum
    float* h1    = Qb;    // Q dead after attn_gemm
    float* fbuf  = Vb;    // attnS dead after output projection
    float* u2    = h;     // h dead after residual add into u1

    const dim3 blk(256);
    const dim3 gMain(DMODEL / 64, MTOT / 128);

    // h = x @ w_in^T + b_in
    gemm_wT<0, false><<<gMain, blk, 0, stream>>>(x, w_in, b_in, nullptr, h, MTOT, INDIM, DMODEL);
    // Q = elu(h@wq^T+bq)+1 ; K = elu(h@wk^T+bk)+1 ; V = h@wv^T+bv
    gemm_wT<1, false><<<gMain, blk, 0, stream>>>(h, wq, bq, nullptr, Qb, MTOT, DMODEL, DMODEL);
    gemm_wT<1, false><<<gMain, blk, 0, stream>>>(h, wk, bk, nullptr, Kb, MTOT, DMODEL, DMODEL);
    gemm_wT<0, false><<<gMain, blk, 0, stream>>>(h, wvv, bvv, nullptr, Vb, MTOT, DMODEL, DMODEL);
    // KV = K^T V (per batch) ; Ksum = sum_l K
    kv_gemm<<<dim3(DMODEL / 64, DMODEL / 128, NB), blk, 0, stream>>>(Kb, Vb, KVb);
    ksum_kernel<<<dim3(DMODEL / 256, NB), blk, 0, stream>>>(Kb, Ksum);
    // Z = 1/(Q . Ksum + eps)
    z_kernel<<<MTOT / 8, blk, 0, stream>>>(Qb, Ksum, Zb);
    // attnS = Z * (Q @ KV)
    attn_gemm<<<dim3(DMODEL / 64, LSEQ / 128, NB), blk, 0, stream>>>(Qb, KVb, Zb, attnS);
    // u1 = attnS @ wo^T + bo + h ; h1 = LN(u1; g1,b1)
    gemm_wT<0, true><<<gMain, blk, 0, stream>>>(attnS, wo, bo, h, u1, MTOT, DMODEL, DMODEL);
    ln_kernel<<<MTOT / 8, blk, 0, stream>>>(u1, g1, b1, h1);
    // f = relu(h1 @ w_ff1^T + b_ff1) ; u2 = f @ w_ff2^T + b_ff2 + h1
    gemm_wT<2, false><<<gMain, blk, 0, stream>>>(h1, w_ff1, b_ff1, nullptr, fbuf, MTOT, DMODEL, DMODEL);
    gemm_wT<0, true><<<gMain, blk, 0, stream>>>(fbuf, w_ff2, b_ff2, h1, u2, MTOT, DMODEL, DMODEL);
    // out[n] = LN(LN(u2[n,0,:];g2,b2);gf,bf) . w_fc + b_fc
    final_kernel<<<NB, 32, 0, stream>>>(u2, g2, b2, gf, bf, w_fc, b_fc, (float*)d_out);
}